// Rasterizer_43344809952125
// MI455X (gfx1250) — compile-verified
//
#include <hip/hip_runtime.h>
#include <math.h>

// ---------------------------------------------------------------------------
// MI455X (gfx1250) rasterizer attribute-interpolation pipeline.
//
// Bandwidth roofline ~45us (@23.3 TB/s). Per-pixel single pass; camera
// transforms on V_WMMA_F32_16X16X4_F32 arranged so D needs no retrieval:
//   A (uniform/wave, hoisted): rows 0-2 = [R | t], rows 3-15 = 0
//   B (per group of 8 pixels): cols 0-7 = [p 1], cols 8-15 = [n 0]
//   => lane n<8 holds pos_cam(pixel n) in c[0..2], lane 8..15 holds raw
//      norm_cam(pixel n-8) in c[0..2]; normalize locally, store directly.
// 4 WMMAs per wave32 (32 pixels). v_rcp/v_rsq fast ops, 32-bit indexing.
// ---------------------------------------------------------------------------

typedef float v2f __attribute__((ext_vector_type(2)));
typedef float v8f __attribute__((ext_vector_type(8)));

#define Bn 4
#define Hn 1024
#define Wn 1024
#define Fn 200000
#define Vn 100000
#define HWn (Hn * Wn)

// Flat float32 output offsets (all < 2^31 => 32-bit index math).
constexpr unsigned O_UV    = 0;                              // (B,H,W,2)
constexpr unsigned O_ALPHA = O_UV    + (unsigned)Bn * HWn * 2;
constexpr unsigned O_FIM   = O_ALPHA + (unsigned)Bn * HWn;
constexpr unsigned O_WMAP  = O_FIM   + (unsigned)Bn * HWn;
constexpr unsigned O_FVIDX = O_WMAP  + (unsigned)Bn * HWn * 3;
constexpr unsigned O_NM    = O_FVIDX + (unsigned)Bn * Fn * 3;
constexpr unsigned O_NMC   = O_NM    + (unsigned)Bn * HWn * 3;
constexpr unsigned O_FV    = O_NMC   + (unsigned)Bn * HWn * 3;
constexpr unsigned O_FVT   = O_FV    + (unsigned)Bn * Fn * 9;
constexpr unsigned O_PM    = O_FVT   + (unsigned)Fn * 6;
constexpr unsigned O_PMC   = O_PM    + (unsigned)Bn * HWn * 3;
constexpr unsigned O_DEPTH = O_PMC   + (unsigned)Bn * HWn * 3;
constexpr unsigned O_VUVZ  = O_DEPTH + (unsigned)Bn * HWn;
constexpr unsigned O_MASK  = O_VUVZ  + (unsigned)Bn * Vn * 3;

// ---------------------------------------------------------------------------
// K0: mesh_span = max over axes of (max - min) of vertices[0]  -> ws[0]
// ---------------------------------------------------------------------------
__global__ __launch_bounds__(1024) void meshspan_kernel(
    const float* __restrict__ verts, float* __restrict__ ws) {
  int tid = threadIdx.x;
  float mn0 =  INFINITY, mn1 =  INFINITY, mn2 =  INFINITY;
  float mx0 = -INFINITY, mx1 = -INFINITY, mx2 = -INFINITY;
  for (unsigned i = tid; i < Vn; i += 1024) {
    float x = verts[i * 3u + 0u], y = verts[i * 3u + 1u], z = verts[i * 3u + 2u];
    mn0 = fminf(mn0, x); mx0 = fmaxf(mx0, x);
    mn1 = fminf(mn1, y); mx1 = fmaxf(mx1, y);
    mn2 = fminf(mn2, z); mx2 = fmaxf(mx2, z);
  }
  #pragma unroll
  for (int off = 16; off > 0; off >>= 1) {
    mn0 = fminf(mn0, __shfl_xor(mn0, off, 32));
    mn1 = fminf(mn1, __shfl_xor(mn1, off, 32));
    mn2 = fminf(mn2, __shfl_xor(mn2, off, 32));
    mx0 = fmaxf(mx0, __shfl_xor(mx0, off, 32));
    mx1 = fmaxf(mx1, __shfl_xor(mx1, off, 32));
    mx2 = fmaxf(mx2, __shfl_xor(mx2, off, 32));
  }
  __shared__ float smn[32][3], smx[32][3];
  int wid = tid >> 5, lane = tid & 31;
  if (lane == 0) {
    smn[wid][0] = mn0; smn[wid][1] = mn1; smn[wid][2] = mn2;
    smx[wid][0] = mx0; smx[wid][1] = mx1; smx[wid][2] = mx2;
  }
  __syncthreads();
  if (wid == 0) {
    mn0 = smn[lane][0]; mn1 = smn[lane][1]; mn2 = smn[lane][2];
    mx0 = smx[lane][0]; mx1 = smx[lane][1]; mx2 = smx[lane][2];
    #pragma unroll
    for (int off = 16; off > 0; off >>= 1) {
      mn0 = fminf(mn0, __shfl_xor(mn0, off, 32));
      mn1 = fminf(mn1, __shfl_xor(mn1, off, 32));
      mn2 = fminf(mn2, __shfl_xor(mn2, off, 32));
      mx0 = fmaxf(mx0, __shfl_xor(mx0, off, 32));
      mx1 = fmaxf(mx1, __shfl_xor(mx1, off, 32));
      mx2 = fmaxf(mx2, __shfl_xor(mx2, off, 32));
    }
    if (lane == 0) {
      ws[0] = fmaxf(fmaxf(mx0 - mn0, mx1 - mn1), mx2 - mn2);
    }
  }
}

// ---------------------------------------------------------------------------
// K1: per (b,v): v_uvz out; bilinear depth[0] sample; b==0 front mask
// ---------------------------------------------------------------------------
__global__ __launch_bounds__(256) void vertex_kernel(
    const float* __restrict__ vuvz_in, const float* __restrict__ depth,
    const float* __restrict__ ws, float* __restrict__ out) {
  unsigned gid = blockIdx.x * 256u + threadIdx.x;
  if (gid >= Bn * Vn) return;
  unsigned b = gid / Vn;
  unsigned v = gid - b * Vn;

  float x = vuvz_in[gid * 3u + 0u];
  float y = vuvz_in[gid * 3u + 1u];
  float z = vuvz_in[gid * 3u + 2u];
  float u  = (x * 0.5f + 0.5f) * (float)Wn;
  float vv = (1.0f - (y * 0.5f + 0.5f)) * (float)Hn;
  out[O_VUVZ + gid * 3u + 0u] = u;
  out[O_VUVZ + gid * 3u + 1u] = vv;
  out[O_VUVZ + gid * 3u + 2u] = z;

  if (b == 0) {
    float x0f = floorf(u), y0f = floorf(vv);
    float tx = u - x0f, ty = vv - y0f;
    int ix0 = min(max((int)x0f, 0), Wn - 1);
    int ix1 = min(ix0 + 1, Wn - 1);
    int iy0 = min(max((int)y0f, 0), Hn - 1);
    int iy1 = min(iy0 + 1, Hn - 1);
    float v00 = depth[(unsigned)iy0 * Wn + (unsigned)ix0];
    float v01 = depth[(unsigned)iy0 * Wn + (unsigned)ix1];
    float v10 = depth[(unsigned)iy1 * Wn + (unsigned)ix0];
    float v11 = depth[(unsigned)iy1 * Wn + (unsigned)ix1];
    float d = v00 * (1.0f - tx) * (1.0f - ty) + v01 * tx * (1.0f - ty) +
              v10 * (1.0f - tx) * ty + v11 * tx * ty;
    float span = ws[0];
    out[O_MASK + v] = (z - d < span * 0.005f) ? 1.0f : 0.0f;
  }
}

// ---------------------------------------------------------------------------
// K2: per (f,k): faces_vt gather; per b: faces_v gather + faces_v_idx
// ---------------------------------------------------------------------------
__global__ __launch_bounds__(256) void faces_kernel(
    const float* __restrict__ verts, const float* __restrict__ texc,
    const int* __restrict__ vt_idx, const int* __restrict__ v_idx,
    float* __restrict__ out) {
  unsigned gid = blockIdx.x * 256u + threadIdx.x;
  if (gid >= Fn * 3u) return;
  unsigned ti = (unsigned)vt_idx[gid];
  out[O_FVT + gid * 2u + 0u] = texc[ti * 2u + 0u];
  out[O_FVT + gid * 2u + 1u] = texc[ti * 2u + 1u];
  #pragma unroll
  for (unsigned b = 0; b < Bn; b++) {
    unsigned vi = (unsigned)v_idx[b * (Fn * 3u) + gid];
    out[O_FVIDX + b * (Fn * 3u) + gid] = (float)vi;
    unsigned o = O_FV + (b * (Fn * 3u) + gid) * 3u;
    out[o + 0u] = verts[vi * 3u + 0u];
    out[o + 1u] = verts[vi * 3u + 1u];
    out[o + 2u] = verts[vi * 3u + 2u];
  }
}

// ---------------------------------------------------------------------------
// K3: per-pixel main pass + WMMA camera transforms.
// Grid exactly covers Bn*HWn (no early-out => EXEC all ones for WMMA).
// HWn % 32 == 0 => a wave never straddles a batch boundary.
// ---------------------------------------------------------------------------
__global__ __launch_bounds__(256) void pixel_kernel(
    const float* __restrict__ verts, const float* __restrict__ texc,
    const float* __restrict__ norms, const float* __restrict__ pose,
    const float* __restrict__ depth, const float* __restrict__ alpha,
    const float* __restrict__ wmapin, const float* __restrict__ fvuvz,
    const int* __restrict__ fim, const int* __restrict__ vt_idx,
    const int* __restrict__ vn_idx, const int* __restrict__ v_idx,
    float* __restrict__ out) {
  unsigned gid = blockIdx.x * 256u + threadIdx.x;  // < Bn*HWn exactly
  unsigned b = gid >> 20;                          // HWn = 1<<20
  unsigned lane = threadIdx.x & 31u;

  unsigned fi = (unsigned)fim[gid];
  float d = depth[gid];

  float u0 = 0.f, u1 = 0.f;
  float nx = 0.f, ny = 0.f, nz = 0.f;
  float px = 0.f, py = 0.f, pz = 0.f;
  float wk[3];
  #pragma unroll
  for (unsigned k = 0; k < 3; k++) {
    unsigned fk = fi * 3u + k;
    float z = fvuvz[(b * (Fn * 3u) + fk) * 3u + 2u];
    float w = __builtin_amdgcn_rcpf(z) * wmapin[gid * 3u + k] * d;
    wk[k] = w;
    unsigned ti = (unsigned)vt_idx[fk];
    u0 += texc[ti * 2u + 0u] * w;
    u1 += texc[ti * 2u + 1u] * w;
    unsigned ni = (unsigned)vn_idx[fk];
    nx += norms[ni * 3u + 0u] * w;
    ny += norms[ni * 3u + 1u] * w;
    nz += norms[ni * 3u + 2u] * w;
    unsigned vi = (unsigned)v_idx[fk];  // faces_v[0][fim]: batch-0 indices
    px += verts[vi * 3u + 0u] * w;
    py += verts[vi * 3u + 1u] * w;
    pz += verts[vi * 3u + 2u] * w;
  }

  // normalize(sum): x / max(||x||, 1e-12) == x * rsq(max(||x||^2, 1e-24))
  float inl = __builtin_amdgcn_rsqf(fmaxf(nx * nx + ny * ny + nz * nz, 1e-24f));
  float nmx = nx * inl, nmy = ny * inl, nmz = nz * inl;

  // Coalesced per-lane outputs
  out[O_UV + gid * 2u + 0u] = u0 - floorf(u0);
  out[O_UV + gid * 2u + 1u] = u1 - floorf(u1);
  out[O_ALPHA + gid] = alpha[gid];
  out[O_FIM + gid] = (float)fi;
  out[O_WMAP + gid * 3u + 0u] = wk[0];
  out[O_WMAP + gid * 3u + 1u] = wk[1];
  out[O_WMAP + gid * 3u + 2u] = wk[2];
  out[O_NM + gid * 3u + 0u] = nmx;
  out[O_NM + gid * 3u + 1u] = nmy;
  out[O_NM + gid * 3u + 2u] = nmz;
  out[O_PM + gid * 3u + 0u] = px;
  out[O_PM + gid * 3u + 1u] = py;
  out[O_PM + gid * 3u + 2u] = pz;
  out[O_DEPTH + gid] = d;

  // ---- WMMA camera transforms: D(16x16) = A(16x4) x B(4x16) + 0 ----
  // A (uniform per wave, hoisted): rows 0-2 = [R | t], rows 3-15 = 0.
  //   A per-lane layout: M = lane&15; lanes 0-15 hold K0(v0),K1(v1);
  //   lanes 16-31 hold K2(v0),K3(v1).
  // B (per 8-pixel group): col n<8  = [p.x p.y p.z 1] of pixel 8g+n,
  //                        col n>=8 = [n.x n.y n.z 0] of pixel 8g+n-8.
  //   B per-lane layout: N = lane&15; same K split as A.
  // D: D[m][n] lives in VGPR m, lane n (n<16). So lane n<8 holds
  //   pos_cam(pixel 8g+n) in c[0..2]; lane 8-15 holds raw norm_cam in
  //   c[0..2] (normalize locally). No retrieval shuffles needed.
  const float* P = pose + b * 12u;  // row-major 3x4: R | t
  unsigned M = lane & 15u;
  bool hiK = lane >= 16u;
  float a0 = 0.f, a1 = 0.f;
  if (M < 3u) {
    a0 = hiK ? P[M * 4u + 2u] : P[M * 4u + 0u];
    a1 = hiK ? P[M * 4u + 3u] : P[M * 4u + 1u];
  }
  v2f Am;
  Am.x = a0;
  Am.y = a1;

  unsigned gw = gid - lane;   // wave's first pixel (same batch, same wave)
  bool nCol = M >= 8u;        // this lane's B column carries a normal

  #pragma unroll
  for (unsigned g = 0; g < 4; g++) {
    int s = (int)(8u * g + (lane & 7u));  // source lane for this column
    float spx = __shfl(px, s, 32), spy = __shfl(py, s, 32), spz = __shfl(pz, s, 32);
    float snx = __shfl(nmx, s, 32), sny = __shfl(nmy, s, 32), snz = __shfl(nmz, s, 32);
    float b0 = !hiK ? (nCol ? snx : spx) : (nCol ? snz : spz);
    float b1 = !hiK ? (nCol ? sny : spy) : (nCol ? 0.0f : 1.0f);
    v2f Bm;
    Bm.x = b0;
    Bm.y = b1;
    v8f c = {};
    c = __builtin_amdgcn_wmma_f32_16x16x4_f32(
        /*neg_a=*/false, Am, /*neg_b=*/false, Bm,
        /*c_mod=*/(short)0, c, /*reuse_a=*/false, /*reuse_b=*/false);
    if (lane < 16u) {
      float x = c[0], y = c[1], z = c[2];
      unsigned pix = gw + 8u * g + (lane & 7u);
      if (lane < 8u) {
        unsigned o = O_PMC + pix * 3u;
        out[o + 0u] = x;
        out[o + 1u] = y;
        out[o + 2u] = z;
      } else {
        float inv = __builtin_amdgcn_rsqf(fmaxf(x * x + y * y + z * z, 1e-24f));
        unsigned o = O_NMC + pix * 3u;
        out[o + 0u] = x * inv;
        out[o + 1u] = y * inv;
        out[o + 2u] = z * inv;
      }
    }
  }
}

// ---------------------------------------------------------------------------
extern "C" void kernel_launch(void* const* d_in, const int* in_sizes, int n_in,
                              void* d_out, int out_size, void* d_ws, size_t ws_size,
                              hipStream_t stream) {
  (void)in_sizes; (void)n_in; (void)out_size; (void)ws_size;
  const float* vertices  = (const float*)d_in[0];
  const float* texcoords = (const float*)d_in[1];
  const float* normals   = (const float*)d_in[2];
  const float* pose      = (const float*)d_in[3];
  const float* depth     = (const float*)d_in[4];
  const float* alpha     = (const float*)d_in[5];
  const float* weightmap = (const float*)d_in[6];
  const float* vuvz      = (const float*)d_in[7];
  const float* fvuvz     = (const float*)d_in[8];
  const int*   fim       = (const int*)d_in[9];
  const int*   vt_idx    = (const int*)d_in[10];
  const int*   vn_idx    = (const int*)d_in[11];
  const int*   v_idx     = (const int*)d_in[12];
  float* out = (float*)d_out;
  float* ws  = (float*)d_ws;

  meshspan_kernel<<<1, 1024, 0, stream>>>(vertices, ws);

  int nbv = (Bn * Vn + 255) / 256;
  vertex_kernel<<<nbv, 256, 0, stream>>>(vuvz, depth, ws, out);

  int nbf = (Fn * 3 + 255) / 256;
  faces_kernel<<<nbf, 256, 0, stream>>>(vertices, texcoords, vt_idx, v_idx, out);

  int nbp = (Bn * HWn) / 256;  // exact: 16384 blocks
  pixel_kernel<<<nbp, 256, 0, stream>>>(vertices, texcoords, normals, pose,
                                        depth, alpha, weightmap, fvuvz,
                                        fim, vt_idx, vn_idx, v_idx, out);
}